// MultiHeadGATLayer_2954937499915
// MI455X (gfx1250) — compile-verified
//
#include <hip/hip_runtime.h>
#include <hip/hip_bf16.h>
#include <math.h>

// Problem constants (match reference).
#define NN 50000
#define DD 64
#define HH 4
#define EE 400000
#define HD (HH * DD)   // 256

typedef float v2f __attribute__((ext_vector_type(2)));
typedef float v8f __attribute__((ext_vector_type(8)));

// ---------------- workspace layout (float offsets) ----------------
// xh      : [N, 256]
// accum   : [N, 256]   (atomic message accumulation; zeroed)
// denom   : [N, 4]     (softmax denominators; zeroed)
// s_src   : [N, 4]
// s_dst   : [N, 4]
// alpha   : [E, 4]     (exp of leaky-relu scores)
// out_raw : [N, 64]    (pre-activation output projection)
#define OFF_XH    ((size_t)0)
#define OFF_ACC   (OFF_XH   + (size_t)NN * HD)
#define OFF_DEN   (OFF_ACC  + (size_t)NN * HD)
#define OFF_SSRC  (OFF_DEN  + (size_t)NN * HH)
#define OFF_SDST  (OFF_SSRC + (size_t)NN * HH)
#define OFF_ALPHA (OFF_SDST + (size_t)NN * HH)
#define OFF_ORAW  (OFF_ALPHA + (size_t)EE * HH)
// zero region = accum + denom (contiguous)
#define ZERO_OFF  OFF_ACC
#define ZERO_CNT  ((size_t)NN * HD + (size_t)NN * HH)

// ---------------- kernel 0: zero scratch ----------------
__global__ void k_zero(float* __restrict__ p, size_t n) {
    size_t i = (size_t)blockIdx.x * blockDim.x + threadIdx.x;
    size_t stride = (size_t)gridDim.x * blockDim.x;
    for (; i < n; i += stride) p[i] = 0.0f;
}

// ---------------- kernel 1: xh = x @ W_lin^T  (WMMA f32 16x16x4) ----------------
// One wave (32 lanes) computes one 16(M) x 16(N) tile, K = 64.
// A 16x4 f32 layout: lanes 0-15 hold row M=lane, K=k..k+1; lanes 16-31 hold K=k+2..k+3.
// B 4x16 f32 layout (mirror): lanes 0-15 hold col N=lane, K=k..k+1; lanes 16-31 K=k+2..k+3.
// C/D: VGPR r -> row r (lanes 0-15) / row r+8 (lanes 16-31), col = lane&15.
__global__ __launch_bounds__(32) void k_gemm_xh(const float* __restrict__ x,      // [N,64]
                                                const float* __restrict__ Wlin,   // [256,64]
                                                float* __restrict__ xh) {         // [N,256]
    const int lane = threadIdx.x;
    const int l    = lane & 15;
    const int half = lane >> 4;
    const int m0   = blockIdx.x * 16;   // N divisible by 16
    const int n0   = blockIdx.y * 16;   // 0..255

    const int arow = m0 + l;
    const int bcol = n0 + l;
    v8f c = {};
    #pragma unroll
    for (int k = 0; k < DD; k += 4) {
        const int ka = k + half * 2;
        const float* ap = x    + (size_t)arow * DD + ka;     // A[m][k]   = x[m][k]
        const float* bp = Wlin + (size_t)bcol * DD + ka;     // B[k][n]   = Wlin[n][k]
        v2f a; a.x = ap[0]; a.y = ap[1];
        v2f b; b.x = bp[0]; b.y = bp[1];
        c = __builtin_amdgcn_wmma_f32_16x16x4_f32(false, a, false, b, (short)0, c,
                                                  false, false);
    }
    const int col = n0 + l;
    #pragma unroll
    for (int r = 0; r < 8; ++r) {
        const int row = m0 + r + half * 8;
        xh[(size_t)row * HD + col] = c[r];
    }
}

// ---------------- kernel 2: per-(node,head) attention scores ----------------
// One wave per (n,h): s = dot(xh[n,h,:], attn[h,:]) over D=64 (2 elems/lane).
__global__ __launch_bounds__(128) void k_scores(const float* __restrict__ xh,
                                                const float* __restrict__ attn_src, // [4,64]
                                                const float* __restrict__ attn_dst, // [4,64]
                                                float* __restrict__ s_src,          // [N,4]
                                                float* __restrict__ s_dst) {
    const int wave = threadIdx.x >> 5;
    const int lane = threadIdx.x & 31;
    const int idx  = blockIdx.x * 4 + wave;          // (n*H + h), N*H divisible by 4
    const int n = idx >> 2;
    const int h = idx & 3;
    if (n >= NN) return;

    const float* xr = xh + (size_t)n * HD + h * DD;
    const float v0 = xr[lane], v1 = xr[lane + 32];
    float ss = v0 * attn_src[h * DD + lane] + v1 * attn_src[h * DD + lane + 32];
    float sd = v0 * attn_dst[h * DD + lane] + v1 * attn_dst[h * DD + lane + 32];
    #pragma unroll
    for (int o = 16; o > 0; o >>= 1) {
        ss += __shfl_xor(ss, o, 32);
        sd += __shfl_xor(sd, o, 32);
    }
    if (lane == 0) {
        s_src[n * HH + h] = ss;
        s_dst[n * HH + h] = sd;
    }
}

// ---------------- kernel 3: per-edge exp(leaky_relu) + denom atomics ----------------
__global__ void k_edge_alpha(const int* __restrict__ ei,   // [2,E] (row0=dst i, row1=src j)
                             const float* __restrict__ s_src,
                             const float* __restrict__ s_dst,
                             float* __restrict__ alpha,    // [E,4]
                             float* __restrict__ denom) {  // [N,4] (zeroed)
    const int e = blockIdx.x * blockDim.x + threadIdx.x;
    if (e >= EE) return;
    const int i = ei[e];
    const int j = ei[EE + e];
    #pragma unroll
    for (int h = 0; h < HH; ++h) {
        float a = s_dst[i * HH + h] + s_src[j * HH + h];
        a = (a > 0.0f) ? a : 0.2f * a;       // leaky_relu(0.2)
        const float w = expf(a);             // raw exp, no max-shift (matches source)
        alpha[(size_t)e * HH + h] = w;
        atomicAdd(&denom[i * HH + h], w);
    }
}

// ---------------- kernel 4: per-edge weighted gather/scatter ----------------
// One 256-thread block per edge: thread t covers channel t of [H*D]=256.
__global__ __launch_bounds__(256) void k_edge_agg(const int* __restrict__ ei,
                                                  const float* __restrict__ xh,
                                                  const float* __restrict__ alpha,
                                                  const float* __restrict__ denom,
                                                  float* __restrict__ accum) { // [N,256] zeroed
    const int e = blockIdx.x;
    const int t = threadIdx.x;        // 0..255
    const int h = t >> 6;
    const int i = ei[e];
    const int j = ei[EE + e];
    const float att = alpha[(size_t)e * HH + h] / (denom[i * HH + h] + 1e-9f);
    const float val = att * xh[(size_t)j * HD + t];
    atomicAdd(&accum[(size_t)i * HD + t], val);
}

// ---------------- kernel 5: out_raw = accum @ W_out^T + b_out (WMMA f32) ----------------
__global__ __launch_bounds__(32) void k_gemm_out(const float* __restrict__ acc,   // [N,256]
                                                 const float* __restrict__ Wout,  // [64,256]
                                                 const float* __restrict__ bout,  // [64]
                                                 float* __restrict__ out_raw) {   // [N,64]
    const int lane = threadIdx.x;
    const int l    = lane & 15;
    const int half = lane >> 4;
    const int m0   = blockIdx.x * 16;
    const int n0   = blockIdx.y * 16;   // 0..63

    const int arow = m0 + l;
    const int bcol = n0 + l;
    v8f c = {};
    #pragma unroll 4
    for (int k = 0; k < HD; k += 4) {
        const int ka = k + half * 2;
        const float* ap = acc  + (size_t)arow * HD + ka;   // A[m][k] = accum[m][k]
        const float* bp = Wout + (size_t)bcol * HD + ka;   // B[k][n] = Wout[n][k]
        v2f a; a.x = ap[0]; a.y = ap[1];
        v2f b; b.x = bp[0]; b.y = bp[1];
        c = __builtin_amdgcn_wmma_f32_16x16x4_f32(false, a, false, b, (short)0, c,
                                                  false, false);
    }
    const int col = n0 + l;
    const float bias = bout[col];
    #pragma unroll
    for (int r = 0; r < 8; ++r) {
        const int row = m0 + r + half * 8;
        out_raw[(size_t)row * DD + col] = c[r] + bias;
    }
}

// ---------------- kernel 6: ELU + residual + LayerNorm ----------------
// One wave per node; 64 channels = 2 per lane; shuffle reductions for mean/var.
__global__ __launch_bounds__(128) void k_elu_ln(const float* __restrict__ out_raw,
                                                const float* __restrict__ x,
                                                const float* __restrict__ ln_g,
                                                const float* __restrict__ ln_b,
                                                float* __restrict__ out) {
    const int wave = threadIdx.x >> 5;
    const int lane = threadIdx.x & 31;
    const int n = blockIdx.x * 4 + wave;   // N divisible by 4
    if (n >= NN) return;

    const float* orow = out_raw + (size_t)n * DD;
    const float* xrow = x + (size_t)n * DD;
    float v0 = orow[lane], v1 = orow[lane + 32];
    v0 = (v0 > 0.0f) ? v0 : (expf(v0) - 1.0f);   // elu(alpha=1)
    v1 = (v1 > 0.0f) ? v1 : (expf(v1) - 1.0f);
    float y0 = v0 + xrow[lane];
    float y1 = v1 + xrow[lane + 32];

    float s = y0 + y1;
    #pragma unroll
    for (int o = 16; o > 0; o >>= 1) s += __shfl_xor(s, o, 32);
    const float mu = s * (1.0f / DD);

    const float d0 = y0 - mu, d1 = y1 - mu;
    float ss = d0 * d0 + d1 * d1;
    #pragma unroll
    for (int o = 16; o > 0; o >>= 1) ss += __shfl_xor(ss, o, 32);
    const float rstd = rsqrtf(ss * (1.0f / DD) + 1e-5f);

    out[(size_t)n * DD + lane]      = d0 * rstd * ln_g[lane]      + ln_b[lane];
    out[(size_t)n * DD + lane + 32] = d1 * rstd * ln_g[lane + 32] + ln_b[lane + 32];
}

// ---------------- host launcher ----------------
extern "C" void kernel_launch(void* const* d_in, const int* in_sizes, int n_in,
                              void* d_out, int out_size, void* d_ws, size_t ws_size,
                              hipStream_t stream) {
    const float* x        = (const float*)d_in[0];
    const int*   ei       = (const int*)d_in[1];     // [2,E]
    const float* Wlin     = (const float*)d_in[2];   // [256,64]
    const float* attn_src = (const float*)d_in[3];   // [4,64]
    const float* attn_dst = (const float*)d_in[4];   // [4,64]
    const float* Wout     = (const float*)d_in[5];   // [64,256]
    const float* bout     = (const float*)d_in[6];   // [64]
    const float* ln_g     = (const float*)d_in[7];   // [64]
    const float* ln_b     = (const float*)d_in[8];   // [64]
    float* out = (float*)d_out;

    float* ws      = (float*)d_ws;
    float* xh      = ws + OFF_XH;
    float* acc     = ws + OFF_ACC;
    float* denom   = ws + OFF_DEN;
    float* s_src   = ws + OFF_SSRC;
    float* s_dst   = ws + OFF_SDST;
    float* alpha   = ws + OFF_ALPHA;
    float* out_raw = ws + OFF_ORAW;

    // 0) zero atomic accumulators
    k_zero<<<4096, 256, 0, stream>>>(ws + ZERO_OFF, ZERO_CNT);

    // 1) xh = x @ Wlin^T   (WMMA f32)
    k_gemm_xh<<<dim3(NN / 16, HD / 16), 32, 0, stream>>>(x, Wlin, xh);

    // 2) per-(node,head) scores
    k_scores<<<(NN * HH) / 4, 128, 0, stream>>>(xh, attn_src, attn_dst, s_src, s_dst);

    // 3) per-edge exp(leaky_relu) + denominators
    k_edge_alpha<<<(EE + 255) / 256, 256, 0, stream>>>(ei, s_src, s_dst, alpha, denom);

    // 4) per-edge aggregation (L2-resident gather + atomic scatter)
    k_edge_agg<<<EE, 256, 0, stream>>>(ei, xh, alpha, denom, acc);

    // 5) out_raw = accum @ Wout^T + b_out  (WMMA f32)
    k_gemm_out<<<dim3(NN / 16, DD / 16), 32, 0, stream>>>(acc, Wout, bout, out_raw);

    // 6) elu + residual + layernorm
    k_elu_ln<<<NN / 4, 128, 0, stream>>>(out_raw, x, ln_g, ln_b, out);
}